// PMSubgoalAttention_90658169684375
// MI455X (gfx1250) — compile-verified
//
#include <hip/hip_runtime.h>
#include <hip/hip_bf16.h>

// Problem constants (must match reference)
#define BB 4096
#define NN 64
#define DD 512
#define HH 512
#define EPS 1e-5f
// bias = exp(-dist^2 / (2*sigma^2)), sigma=0.25 -> 1/(2*sigma^2) = 8
#define INV_2SIG2 8.0f

typedef __attribute__((ext_vector_type(2))) float v2f;
typedef __attribute__((ext_vector_type(8))) float v8f;

// ---------------------------------------------------------------------------
// Kernel 1: h = pm*Wq1 + bq1 ; LayerNorm(H) ; ReLU  -> relu_h (B,H)
// One block (256 threads) per batch row; each thread owns 2 of 512 elements.
// ---------------------------------------------------------------------------
__global__ void k_query_mlp_ln(const float* __restrict__ pm,
                               const float* __restrict__ Wq1,
                               const float* __restrict__ bq1,
                               const float* __restrict__ ln_g,
                               const float* __restrict__ ln_b,
                               float* __restrict__ relu_h) {
  const int b = blockIdx.x;
  const int t = threadIdx.x;           // 0..255
  __shared__ float red[256];

  const float p = pm[b];               // PM == 1
  const float h0 = p * Wq1[t]       + bq1[t];
  const float h1 = p * Wq1[t + 256] + bq1[t + 256];

  // mean
  red[t] = h0 + h1;
  __syncthreads();
  for (int off = 128; off > 0; off >>= 1) {
    if (t < off) red[t] += red[t + off];
    __syncthreads();
  }
  const float mean = red[0] * (1.0f / 512.0f);
  __syncthreads();

  // variance (biased, like jnp.var)
  const float d0 = h0 - mean, d1 = h1 - mean;
  red[t] = d0 * d0 + d1 * d1;
  __syncthreads();
  for (int off = 128; off > 0; off >>= 1) {
    if (t < off) red[t] += red[t + off];
    __syncthreads();
  }
  const float rstd = rsqrtf(red[0] * (1.0f / 512.0f) + EPS);

  const float y0 = d0 * rstd * ln_g[t]       + ln_b[t];
  const float y1 = d1 * rstd * ln_g[t + 256] + ln_b[t + 256];
  relu_h[(size_t)b * HH + t]       = fmaxf(y0, 0.0f);
  relu_h[(size_t)b * HH + t + 256] = fmaxf(y1, 0.0f);
}

// ---------------------------------------------------------------------------
// WMMA fp32 GEMM:  C(M,N) = A(M,K) @ op(B) + bias
//   BT=true : op(B) = B^T with B stored (N,K)  row-major   (x @ W.T)
//   BT=false: op(B) = B   with B stored (K,N)  row-major   (x @ W)
// Block: 256 threads = 8 waves, wave grid 4x2 -> 64x32 C tile per block.
// Each wave computes one 16x16 tile with V_WMMA_F32_16X16X4_F32.
// Fragments for the whole 32-wide K chunk are prefetched into registers
// before the 8 back-to-back WMMAs (batches DS loads behind one dscnt wait).
// ---------------------------------------------------------------------------
template <bool BT>
__global__ void k_gemm_wmma(const float* __restrict__ A,
                            const float* __restrict__ Bm,
                            const float* __restrict__ bias,   // may be null
                            float* __restrict__ Cout,
                            int M, int Nc, int K) {
  const int tid  = threadIdx.x;
  const int lane = tid & 31;
  const int wid  = tid >> 5;          // 0..7
  const int wm   = wid & 3;           // 0..3 : M sub-tile
  const int wn   = wid >> 2;          // 0..1 : N sub-tile
  const int bm   = blockIdx.x * 64;
  const int bn   = blockIdx.y * 32;

  __shared__ float Alds[64][33];      // [m][k], padded
  __shared__ float Blds[32][33];      // [k][n], padded

  v8f c = {};

  const int mRow = wm * 16 + (lane & 15);
  const int nCol = wn * 16 + (lane & 15);
  const int kOff = (lane >> 4) * 2;   // 0 for lanes 0-15, 2 for 16-31

  for (int kc = 0; kc < K; kc += 32) {
    // Stage A tile (64 x 32), coalesced along k
#pragma unroll
    for (int j = 0; j < 8; ++j) {
      const int idx = tid + j * 256;
      const int r = idx >> 5, cc = idx & 31;
      Alds[r][cc] = A[(size_t)(bm + r) * K + kc + cc];
    }
    // Stage B tile (32k x 32n)
#pragma unroll
    for (int j = 0; j < 4; ++j) {
      const int idx = tid + j * 256;
      if (BT) {
        const int n = idx >> 5, k = idx & 31;
        Blds[k][n] = Bm[(size_t)(bn + n) * K + kc + k];
      } else {
        const int k = idx >> 5, n = idx & 31;
        Blds[k][n] = Bm[(size_t)(kc + k) * Nc + bn + n];
      }
    }
    __syncthreads();

    // Prefetch all 8 fragment pairs, then run WMMAs back-to-back.
    v2f af[8], bf[8];
#pragma unroll
    for (int s = 0; s < 8; ++s) {
      const int k0 = s * 4;
      af[s].x = Alds[mRow][k0 + kOff];
      af[s].y = Alds[mRow][k0 + kOff + 1];
      bf[s].x = Blds[k0 + kOff][nCol];
      bf[s].y = Blds[k0 + kOff + 1][nCol];
    }
#pragma unroll
    for (int s = 0; s < 8; ++s) {
      c = __builtin_amdgcn_wmma_f32_16x16x4_f32(
              /*neg_a=*/false, af[s], /*neg_b=*/false, bf[s],
              /*c_mod=*/(short)0, c, /*reuse_a=*/false, /*reuse_b=*/false);
    }
    __syncthreads();
  }

  const int gn  = bn + wn * 16 + (lane & 15);
  const int gm0 = bm + wm * 16 + ((lane >> 4) << 3);
  const float bb = bias ? bias[gn] : 0.0f;
#pragma unroll
  for (int r = 0; r < 8; ++r)
    Cout[(size_t)(gm0 + r) * Nc + gn] = c[r] + bb;
}

// ---------------------------------------------------------------------------
// qdot[b] = query[b,:] . bk   (one wave per batch row, 8 rows per block)
// ---------------------------------------------------------------------------
__global__ void k_dot_bk(const float* __restrict__ query,
                         const float* __restrict__ bk,
                         float* __restrict__ qdot) {
  const int lane = threadIdx.x & 31;
  const int b = blockIdx.x * 8 + (threadIdx.x >> 5);
  const float* q = query + (size_t)b * HH;
  float s = 0.0f;
#pragma unroll
  for (int i = 0; i < 16; ++i) {
    const int j = lane + 32 * i;
    s += q[j] * bk[j];
  }
#pragma unroll
  for (int off = 16; off > 0; off >>= 1) s += __shfl_down(s, off, 32);
  if (lane == 0) qdot[b] = s;
}

// ---------------------------------------------------------------------------
// Kernel 4 (the HBM-bound one): per batch b
//   scores[n] = scale*(subgoals[b,n,:].qk[b,:] + qdot[b]) + gauss_bias(n)
//   mask -> softmax -> attn (written to output)
//   s[b,d] = sum_n attn[n]*subgoals[b,n,d]
// subgoals[b] (128 KB) is read twice; the second pass hits L2 (192 MB).
// One block (256 threads = 8 waves) per batch.
// ---------------------------------------------------------------------------
__global__ void k_attention(const float* __restrict__ subgoals,
                            const float* __restrict__ qk,
                            const float* __restrict__ qdot,
                            const float* __restrict__ pm,
                            const int*   __restrict__ mask,
                            float* __restrict__ attn_out,
                            float* __restrict__ s_out) {
  const int b   = blockIdx.x;
  const int tid = threadIdx.x;       // 0..255
  const int lane = tid & 31;
  const int wid  = tid >> 5;         // 0..7

  __shared__ float qk_s[512];
  __shared__ float sc[64];
  __shared__ float attn_s[64];

  *(float2*)(qk_s + 2 * tid) = *(const float2*)(qk + (size_t)b * DD + 2 * tid);
  __syncthreads();

  const float* sg = subgoals + (size_t)b * NN * DD;
  const float scale = 0.044194173824159216f;   // 512^-0.5

  // Pass 1: dot(subgoals[b,n,:], qk[b,:]) ; wave w owns rows w*8..w*8+7
#pragma unroll
  for (int rr = 0; rr < 8; ++rr) {
    const int n = wid * 8 + rr;
    const float* row = sg + (size_t)n * DD;
    float acc = 0.0f;
#pragma unroll
    for (int i = 0; i < 4; ++i) {
      const float4 v = *(const float4*)(row + i * 128 + lane * 4);
      const float4 q = *(const float4*)(qk_s + i * 128 + lane * 4);
      acc += v.x * q.x + v.y * q.y + v.z * q.z + v.w * q.w;
    }
#pragma unroll
    for (int off = 16; off > 0; off >>= 1) acc += __shfl_down(acc, off, 32);
    if (lane == 0) sc[n] = acc;
  }
  __syncthreads();

  // Softmax over N=64 by wave 0 (2 values per lane)
  if (wid == 0) {
    const float qd = qdot[b];
    const float p  = pm[b];
    float s0, s1;
    {
      const int n0 = lane, n1 = lane + 32;
      const float d0v = p - (float)n0 * (1.0f / 63.0f);
      const float d1v = p - (float)n1 * (1.0f / 63.0f);
      s0 = scale * (sc[n0] + qd) + expf(-d0v * d0v * INV_2SIG2);
      s1 = scale * (sc[n1] + qd) + expf(-d1v * d1v * INV_2SIG2);
      if (mask[b * NN + n0] == 0) s0 = -1e9f;
      if (mask[b * NN + n1] == 0) s1 = -1e9f;
    }
    float m = fmaxf(s0, s1);
#pragma unroll
    for (int off = 16; off > 0; off >>= 1) m = fmaxf(m, __shfl_xor(m, off, 32));
    const float e0 = expf(s0 - m);
    const float e1 = expf(s1 - m);
    float sum = e0 + e1;
#pragma unroll
    for (int off = 16; off > 0; off >>= 1) sum += __shfl_xor(sum, off, 32);
    const float inv = 1.0f / sum;
    const float a0 = e0 * inv, a1 = e1 * inv;
    attn_s[lane]      = a0;
    attn_s[lane + 32] = a1;
    attn_out[(size_t)b * NN + lane]      = a0;
    attn_out[(size_t)b * NN + lane + 32] = a1;
  }
  __syncthreads();

  // Pass 2: s[d] = sum_n attn[n] * subgoals[b,n,d]  (L2 hits), float2/lane
  float2 acc = {0.0f, 0.0f};
#pragma unroll 4
  for (int n = 0; n < NN; ++n) {
    const float a = attn_s[n];
    const float2 v = *(const float2*)(sg + (size_t)n * DD + 2 * tid);
    acc.x += a * v.x;
    acc.y += a * v.y;
  }
  *(float2*)(s_out + (size_t)b * DD + 2 * tid) = acc;
}

// ---------------------------------------------------------------------------
// Host launcher
// ---------------------------------------------------------------------------
extern "C" void kernel_launch(void* const* d_in, const int* in_sizes, int n_in,
                              void* d_out, int out_size, void* d_ws, size_t ws_size,
                              hipStream_t stream) {
  (void)in_sizes; (void)n_in; (void)out_size; (void)ws_size;

  const float* pm   = (const float*)d_in[0];
  const float* subg = (const float*)d_in[1];
  const int*   mask = (const int*)  d_in[2];
  const float* Wq1  = (const float*)d_in[3];
  const float* bq1  = (const float*)d_in[4];
  const float* ln_g = (const float*)d_in[5];
  const float* ln_b = (const float*)d_in[6];
  const float* Wq2  = (const float*)d_in[7];
  const float* bq2  = (const float*)d_in[8];
  const float* Wk   = (const float*)d_in[9];
  const float* bk   = (const float*)d_in[10];
  const float* Wv   = (const float*)d_in[11];
  const float* bv   = (const float*)d_in[12];

  float* out      = (float*)d_out;                 // (B,H)
  float* attn_out = out + (size_t)BB * HH;         // (B,N)

  // Workspace: relu_h (reused as s), query, qk, qdot  -> ~25.2 MB
  float* ws     = (float*)d_ws;
  float* relu_h = ws;                              // B*H, dead after GEMM 1
  float* query  = ws + (size_t)1 * BB * HH;
  float* qk     = ws + (size_t)2 * BB * HH;
  float* qdot   = ws + (size_t)3 * BB * HH;        // B floats
  float* s_buf  = relu_h;                          // reuse

  // 1) query MLP first layer + LN + ReLU
  k_query_mlp_ln<<<dim3(BB), dim3(256), 0, stream>>>(pm, Wq1, bq1, ln_g, ln_b, relu_h);

  // 2) query = relu_h @ Wq2^T + bq2   (NT)
  k_gemm_wmma<true><<<dim3(BB / 64, HH / 32), dim3(256), 0, stream>>>(
      relu_h, Wq2, bq2, query, BB, HH, HH);

  // 3) qk = query @ Wk   (NN) ;  qdot = query . bk
  k_gemm_wmma<false><<<dim3(BB / 64, DD / 32), dim3(256), 0, stream>>>(
      query, Wk, nullptr, qk, BB, DD, HH);
  k_dot_bk<<<dim3(BB / 8), dim3(256), 0, stream>>>(query, bk, qdot);

  // 4) scores/softmax/attn + weighted subgoal sum  (HBM-bound stream)
  k_attention<<<dim3(BB), dim3(256), 0, stream>>>(
      subg, qk, qdot, pm, mask, attn_out, s_buf);

  // 5) out = s @ Wv^T + bv   (NT)
  k_gemm_wmma<true><<<dim3(BB / 64, HH / 32), dim3(256), 0, stream>>>(
      s_buf, Wv, bv, out, BB, HH, HH);
}